// NN_ion_52347061403910
// MI455X (gfx1250) — compile-verified
//
#include <hip/hip_runtime.h>
#include <math.h>

// CDNA5 (gfx1250, wave32) implementation.
// f32 WMMA 16x16x4 used for the 16x16 hidden layer (6 jet RHS x 2 bases)
// and the 32x32 E-net layer (2 N-tiles x 8 K-chunks).

typedef __attribute__((ext_vector_type(2))) float v2f;
typedef __attribute__((ext_vector_type(8))) float v8f;

#define WAVES 4
#define PTS_PER_BLOCK (WAVES * 16)

__device__ __forceinline__ float sigm(float u) { return 1.0f / (1.0f + expf(-u)); }

__device__ __forceinline__ v8f splat8(float v) {
  v8f r;
#pragma unroll
  for (int i = 0; i < 8; ++i) r[i] = v;
  return r;
}

// D(16x16) += A(16x16, LDS row-major [m][k]) * B, where B[k][n] = Wr[n*16+k]
// (i.e. Wr is the weight matrix row-major [out][in], exactly W_H2's layout).
// ISA 7.12.2 f32 layouts: A: lane L -> M=L%16, regs hold K = k0+2*(L/16)+{0,1}
//                         B: lane L -> N=L%16, regs hold same K pair
//                         D: lane L, reg r -> M = r + 8*(L/16), N = L%16
__device__ __forceinline__ v8f wmma16(const float* __restrict__ A,
                                      const float* __restrict__ Wr,
                                      v8f acc, int lane) {
  const int half2 = (lane >> 4) << 1;
  const int mn = lane & 15;
#pragma unroll
  for (int k0 = 0; k0 < 16; k0 += 4) {
    const int ka = k0 + half2;
    v2f a, b;
    a[0] = A[mn * 16 + ka];
    a[1] = A[mn * 16 + ka + 1];
    b[0] = Wr[mn * 16 + ka];
    b[1] = Wr[mn * 16 + ka + 1];
    acc = __builtin_amdgcn_wmma_f32_16x16x4_f32(false, a, false, b, (short)0, acc,
                                                false, false);
  }
  return acc;
}

// D(16x16 tile ntile) += A(16x32 LDS) * W_E2^T tile, Wr row-major [32][32]
__device__ __forceinline__ v8f wmma32(const float* __restrict__ A,
                                      const float* __restrict__ Wr, int ntile,
                                      v8f acc, int lane) {
  const int half2 = (lane >> 4) << 1;
  const int mn = lane & 15;
  const int nn = mn + 16 * ntile;
#pragma unroll
  for (int k0 = 0; k0 < 32; k0 += 4) {
    const int ka = k0 + half2;
    v2f a, b;
    a[0] = A[mn * 32 + ka];
    a[1] = A[mn * 32 + ka + 1];
    b[0] = Wr[nn * 32 + ka];
    b[1] = Wr[nn * 32 + ka + 1];
    acc = __builtin_amdgcn_wmma_f32_16x16x4_f32(false, a, false, b, (short)0, acc,
                                                false, false);
  }
  return acc;
}

// Reduce partials over the N dimension (16 lanes of each half-wave); lanes with
// (lane&15)==0 hold sums for 8 points and scatter them to dst16[point].
__device__ __forceinline__ void reduceP(float p[8], float* dst16, int lane) {
#pragma unroll
  for (int off = 1; off < 16; off <<= 1)
#pragma unroll
    for (int r = 0; r < 8; ++r) p[r] += __shfl_xor(p[r], off, 32);
  if ((lane & 15) == 0) {
    const int base = (lane >> 4) * 8;
#pragma unroll
    for (int r = 0; r < 8; ++r) dst16[base + r] = p[r];
  }
}

__device__ __forceinline__ void reduceN(v8f d, float w, float* dst16, int lane) {
  float p[8];
#pragma unroll
  for (int r = 0; r < 8; ++r) p[r] = d[r] * w;
  reduceP(p, dst16, lane);
}

__global__ __launch_bounds__(128) void h2ion_main(
    const float* __restrict__ x, const float* __restrict__ y,
    const float* __restrict__ z, const float* __restrict__ R,
    const float* __restrict__ W_H1, const float* __restrict__ b_H1,
    const float* __restrict__ W_H2, const float* __restrict__ b_H2,
    const float* __restrict__ W_out, const float* __restrict__ b_out,
    const float* __restrict__ W_E1, const float* __restrict__ b_E1,
    const float* __restrict__ W_E2, const float* __restrict__ b_E2,
    const float* __restrict__ W_Eout, const float* __restrict__ b_Eout,
    const float* __restrict__ W_DL, const float* __restrict__ b_DL,
    const float* __restrict__ W_D, const float* __restrict__ b_D,
    float* __restrict__ psiArr, float* __restrict__ accum, int n) {
  __shared__ float sW_H2[256];
  __shared__ float sW_E2[1024];
  __shared__ float sW_out[16];
  __shared__ float sW_Eout[32];
  __shared__ float sA[WAVES][6][256];   // 6 jet A-matrices per wave (16pts x 16k)
  __shared__ float sAe[WAVES][512];     // E-net activations (16pts x 32k)
  __shared__ float sS[WAVES][7][16];    // reduced per-point scalars

  const int tid = threadIdx.x;
  for (int i = tid; i < 256; i += blockDim.x) sW_H2[i] = W_H2[i];
  for (int i = tid; i < 1024; i += blockDim.x) sW_E2[i] = W_E2[i];
  if (tid < 16) sW_out[tid] = W_out[tid];
  if (tid < 32) sW_Eout[tid] = W_Eout[tid];
  __syncthreads();

  const int lane = tid & 31;
  const int wave = tid >> 5;
  const int mn = lane & 15;
  const int p = mn;  // point slot owned by this lane (lanes 16-31 mirror)
  const bool owner = (lane < 16);
  const int pidx = blockIdx.x * PTS_PER_BLOCK + wave * 16 + p;
  const bool valid = (pidx < n);
  const int pc = valid ? pidx : (n - 1);

  const float xi = x[pc], yi = y[pc], zi = z[pc], Ri = R[pc];
  const float yz = yi * yi + zi * zi;
  const float r1 = sqrtf((xi - Ri) * (xi - Ri) + yz);
  const float r2 = sqrtf((xi + Ri) * (xi + Ri) + yz);
  const float f1 = expf(-r1), f2 = expf(-r2);
  const float cth = (xi * xi - Ri * Ri + yz) / (r1 * r2);  // grad r1 . grad r2

  const v8f vzero = splat8(0.f);
  v8f aV = vzero, aA = vzero, aB = vzero, aAA = vzero, aAB = vzero, aBB = vzero;
  const float bh2 = b_H2[mn];

#pragma unroll
  for (int bse = 0; bse < 2; ++bse) {
    // base(f1,f2) and base(f2,f1)  (f1m==f2, f2m==f1 by symmetry)
    const float ain = bse ? f2 : f1;
    const float bin = bse ? f1 : f2;
    if (owner) {
#pragma unroll
      for (int j = 0; j < 16; ++j) {
        const float w0 = W_H1[2 * j], w1 = W_H1[2 * j + 1];
        const float v = sigm(w0 * ain + w1 * bin + b_H1[j]);
        const float s1 = v * (1.f - v);
        const float s2 = s1 * (1.f - 2.f * v);
        sA[wave][0][p * 16 + j] = v;
        sA[wave][1][p * 16 + j] = s1 * w0;
        sA[wave][2][p * 16 + j] = s1 * w1;
        sA[wave][3][p * 16 + j] = s2 * w0 * w0;
        sA[wave][4][p * 16 + j] = s2 * w0 * w1;
        sA[wave][5][p * 16 + j] = s2 * w1 * w1;
      }
    }
    __syncthreads();

    v8f Uv = splat8(bh2);
    Uv = wmma16(sA[wave][0], sW_H2, Uv, lane);
    v8f Uda = wmma16(sA[wave][1], sW_H2, vzero, lane);
    v8f Udb = wmma16(sA[wave][2], sW_H2, vzero, lane);
    v8f Udaa = wmma16(sA[wave][3], sW_H2, vzero, lane);
    v8f Udab = wmma16(sA[wave][4], sW_H2, vzero, lane);
    v8f Udbb = wmma16(sA[wave][5], sW_H2, vzero, lane);

    v8f hv, hda, hdb, hdaa, hdab, hdbb;
#pragma unroll
    for (int r = 0; r < 8; ++r) {
      const float s = sigm(Uv[r]);
      const float s1 = s * (1.f - s);
      const float s2 = s1 * (1.f - 2.f * s);
      hv[r] = s;
      hda[r] = s1 * Uda[r];
      hdb[r] = s1 * Udb[r];
      hdaa[r] = s2 * Uda[r] * Uda[r] + s1 * Udaa[r];
      hdab[r] = s2 * Uda[r] * Udb[r] + s1 * Udab[r];
      hdbb[r] = s2 * Udb[r] * Udb[r] + s1 * Udbb[r];
    }
    if (bse == 0) {
      aV += hv; aA += hda; aB += hdb; aAA += hdaa; aAB += hdab; aBB += hdbb;
    } else {  // base(f2,f1): d/df1 = d/db', d/df2 = d/da'
      aV += hv; aA += hdb; aB += hda; aAA += hdbb; aAB += hdab; aBB += hdaa;
    }
    __syncthreads();
  }

  // E network: e1 = sigmoid(W_E1*R + b_E1); layer2 via WMMA (16x32 @ 32x32)
  if (owner) {
#pragma unroll
    for (int j = 0; j < 32; ++j)
      sAe[wave][p * 32 + j] = sigm(W_E1[j] * Ri + b_E1[j]);
  }
  __syncthreads();
  v8f Ue0 = splat8(b_E2[mn]);
  v8f Ue1 = splat8(b_E2[mn + 16]);
  Ue0 = wmma32(sAe[wave], sW_E2, 0, Ue0, lane);
  Ue1 = wmma32(sAe[wave], sW_E2, 1, Ue1, lane);
  float pE[8];
#pragma unroll
  for (int r = 0; r < 8; ++r)
    pE[r] = sigm(Ue0[r]) * sW_Eout[mn] + sigm(Ue1[r]) * sW_Eout[mn + 16];

  // Output-layer contractions (sum over N with W_out / W_Eout weights)
  reduceN(aV, sW_out[mn], sS[wave][0], lane);
  reduceN(aA, sW_out[mn], sS[wave][1], lane);
  reduceN(aB, sW_out[mn], sS[wave][2], lane);
  reduceN(aAA, sW_out[mn], sS[wave][3], lane);
  reduceN(aAB, sW_out[mn], sS[wave][4], lane);
  reduceN(aBB, sW_out[mn], sS[wave][5], lane);
  reduceP(pE, sS[wave][6], lane);
  __syncthreads();

  const float Sv = sS[wave][0][p];
  const float S1 = sS[wave][1][p];
  const float S2 = sS[wave][2][p];
  const float S11 = sS[wave][3][p];
  const float S12 = sS[wave][4][p];
  const float S22 = sS[wave][5][p];
  const float Ei = sS[wave][6][p] + b_Eout[0];

  float dec = b_D[0];
#pragma unroll
  for (int j = 0; j < 10; ++j) dec += W_D[j] * sigm(W_DL[j] * Ri + b_DL[j]);

  const float psi = (Sv + b_out[0]) * dec + f1 + f2;
  const float L1 = f1 * (1.f - 2.f / r1);  // lap of exp(-r1)
  const float L2 = f2 * (1.f - 2.f / r2);
  const float lap = dec * (S11 * f1 * f1 + 2.f * S12 * f1 * f2 * cth +
                           S22 * f2 * f2 + S1 * L1 + S2 * L2) +
                    L1 + L2;
  const float ham = -0.5f * lap - (1.f / r1 + 1.f / r2) * psi;
  const float res = ham - Ei * psi;

  if (owner && valid) psiArr[pidx] = psi;
  float contrib = (owner && valid) ? res * res : 0.f;
#pragma unroll
  for (int off = 1; off < 32; off <<= 1) contrib += __shfl_xor(contrib, off, 32);
  if (lane == 0) atomicAdd(accum, contrib);
}

__global__ void h2ion_init(float* ws) {
  ws[0] = 0.f;
  ws[1] = 0.f;
  ws[2] = 0.f;
}

__global__ void h2ion_boundary(const float* __restrict__ psiArr,
                               const int* __restrict__ b1,
                               const int* __restrict__ b2,
                               float* __restrict__ ws, int nb) {
  const int i = blockIdx.x * blockDim.x + threadIdx.x;
  float v1 = 0.f, v2 = 0.f;
  if (i < nb) {
    const float p1 = psiArr[b1[i]];
    const float p2 = psiArr[b2[i]];
    v1 = p1 * p1;
    v2 = p2 * p2;
  }
#pragma unroll
  for (int off = 1; off < 32; off <<= 1) {
    v1 += __shfl_xor(v1, off, 32);
    v2 += __shfl_xor(v2, off, 32);
  }
  if ((threadIdx.x & 31) == 0) {
    atomicAdd(&ws[1], v1);
    atomicAdd(&ws[2], v2);
  }
}

__global__ void h2ion_final(const float* ws, float* out, int n, int nb) {
  out[0] = ws[0] / (float)n + ws[1] / (float)nb + ws[2] / (float)nb;
}

extern "C" void kernel_launch(void* const* d_in, const int* in_sizes, int n_in,
                              void* d_out, int out_size, void* d_ws,
                              size_t ws_size, hipStream_t stream) {
  const float* x = (const float*)d_in[0];
  const float* y = (const float*)d_in[1];
  const float* z = (const float*)d_in[2];
  const float* R = (const float*)d_in[3];
  const int* b1 = (const int*)d_in[4];
  const int* b2 = (const int*)d_in[5];
  const float* W_H1 = (const float*)d_in[6];
  const float* b_H1 = (const float*)d_in[7];
  const float* W_H2 = (const float*)d_in[8];
  const float* b_H2 = (const float*)d_in[9];
  const float* W_out = (const float*)d_in[10];
  const float* b_out = (const float*)d_in[11];
  const float* W_E1 = (const float*)d_in[12];
  const float* b_E1 = (const float*)d_in[13];
  const float* W_E2 = (const float*)d_in[14];
  const float* b_E2 = (const float*)d_in[15];
  const float* W_Eout = (const float*)d_in[16];
  const float* b_Eout = (const float*)d_in[17];
  const float* W_DL = (const float*)d_in[18];
  const float* b_DL = (const float*)d_in[19];
  const float* W_D = (const float*)d_in[20];
  const float* b_D = (const float*)d_in[21];

  const int n = in_sizes[0];
  const int nb = in_sizes[4];

  float* wsf = (float*)d_ws;
  float* psiArr = wsf + 16;  // [n] floats of scratch for psi

  h2ion_init<<<1, 1, 0, stream>>>(wsf);

  const int blocks = (n + PTS_PER_BLOCK - 1) / PTS_PER_BLOCK;
  h2ion_main<<<blocks, WAVES * 32, 0, stream>>>(
      x, y, z, R, W_H1, b_H1, W_H2, b_H2, W_out, b_out, W_E1, b_E1, W_E2, b_E2,
      W_Eout, b_Eout, W_DL, b_DL, W_D, b_D, psiArr, wsf, n);

  h2ion_boundary<<<(nb + 255) / 256, 256, 0, stream>>>(psiArr, b1, b2, wsf, nb);
  h2ion_final<<<1, 1, 0, stream>>>(wsf, (float*)d_out, n, nb);
}